// Rules_67619965108887
// MI455X (gfx1250) — compile-verified
//
#include <hip/hip_runtime.h>

// fired[b,s,r] = prod over selected MFs of x[b,s,f]
// Computed in log2 domain as a WMMA f32 GEMM:
//   fired = exp2( L @ rulesT ),  L = (x==0 ? 0 : log2(x))
// Shapes: L (32768 x 15 -> pad K=16), rulesT (15 x 125 -> pad 16 x 128).

typedef __attribute__((ext_vector_type(2))) float v2f;
typedef __attribute__((ext_vector_type(8))) float v8f;

#define NB   16
#define NS   2048
#define NF   15
#define NR   125
#define KPAD 16
#define NPAD 128          // 8 N-tiles of 16
#define NTILES 8
#define BSPOS (NB * NS)   // 32768 positions
#define MTILES (BSPOS / 16)  // 2048

__global__ __launch_bounds__(256) void fired_wmma_kernel(
    const float* __restrict__ x,      // (BSPOS, 15)
    const float* __restrict__ rules,  // (125, 15) one-hot mask
    float* __restrict__ out)          // (BSPOS, 125)
{
    // ---- Stage padded B^T (K x N) = rules^T into LDS once per workgroup ----
    __shared__ float Bs[KPAD][NPAD];  // 8 KB
    const int t = threadIdx.x;
    for (int i = t; i < KPAD * NPAD; i += 256) {
        const int k = i / NPAD, n = i % NPAD;
        float v = 0.0f;
        if (k < NF && n < NR) v = rules[n * NF + k];  // transpose on the fly
        Bs[k][n] = v;
    }
    __syncthreads();

    const int lane = t & 31;
    const int half = lane >> 4;   // which 16-lane half of the wave
    const int m    = lane & 15;   // row (A) / column (B,D) index within tile

    // ---- Hoist all B fragments into registers (shared by every M-tile) ----
    // 16x16x4 f32 B layout (4xN, mirrors A): VGPR0 = {K=2h}, VGPR1 = {K=2h+1},
    // lanes 0-15 -> N=0..15 (half h=0 holds K=0,1; h=1 holds K=2,3 per step).
    v2f bfrag[NTILES][4];
    #pragma unroll
    for (int nt = 0; nt < NTILES; ++nt) {
        #pragma unroll
        for (int s = 0; s < 4; ++s) {
            const int k0 = 4 * s + 2 * half;
            bfrag[nt][s].x = Bs[k0][nt * 16 + m];
            bfrag[nt][s].y = Bs[k0 + 1][nt * 16 + m];
        }
    }

    const int wave       = (int)((blockIdx.x * blockDim.x + t) >> 5);
    const int wavesTotal = (int)((gridDim.x * blockDim.x) >> 5);

    for (int mt = wave; mt < MTILES; mt += wavesTotal) {
        const int posBase = mt * 16;
        const float* __restrict__ xrow = x + (size_t)(posBase + m) * NF;

        // ---- Build A fragment: log2(x) with zero -> multiplicative identity ----
        // 16x4 f32 A layout: lanes 0-15 -> M, VGPR0 = K=2h, VGPR1 = K=2h+1.
        v2f afrag[4];
        #pragma unroll
        for (int s = 0; s < 4; ++s) {
            const int k0 = 4 * s + 2 * half;
            const float x0 = xrow[k0];
            afrag[s].x = (x0 == 0.0f) ? 0.0f : __builtin_amdgcn_logf(x0);
            if (k0 + 1 < NF) {
                const float x1 = xrow[k0 + 1];
                afrag[s].y = (x1 == 0.0f) ? 0.0f : __builtin_amdgcn_logf(x1);
            } else {
                afrag[s].y = 0.0f;  // K=15 zero-pad
            }
        }

        float* __restrict__ orow = out + (size_t)posBase * NR;
        const int n = m;  // D-matrix column this lane owns within the N-tile

        #pragma unroll
        for (int nt = 0; nt < NTILES; ++nt) {
            v8f c = {};
            #pragma unroll
            for (int s = 0; s < 4; ++s) {
                // D = A(16x4) * B(4x16) + C ; accumulates sum of log2 factors
                c = __builtin_amdgcn_wmma_f32_16x16x4_f32(
                        /*neg_a=*/false, afrag[s],
                        /*neg_b=*/false, bfrag[nt][s],
                        /*c_mod=*/(short)0, c,
                        /*reuse_a=*/false, /*reuse_b=*/false);
            }
            const int ncol = nt * 16 + n;
            if (ncol < NR) {  // predicate off the 3 pad columns of the last tile
                #pragma unroll
                for (int v = 0; v < 8; ++v) {
                    const int row = v + 8 * half;  // D layout: VGPR v -> M=v (+8 for hi half)
                    orow[(size_t)row * NR + ncol] = __builtin_amdgcn_exp2f(c[v]);
                }
            }
        }
    }
}

extern "C" void kernel_launch(void* const* d_in, const int* in_sizes, int n_in,
                              void* d_out, int out_size, void* d_ws, size_t ws_size,
                              hipStream_t stream) {
    const float* x     = (const float*)d_in[0];  // (16, 2048, 15) f32
    const float* rules = (const float*)d_in[1];  // (125, 15) f32
    (void)in_sizes; (void)n_in; (void)d_ws; (void)ws_size; (void)out_size;
    float* out = (float*)d_out;                  // (16, 2048, 125) f32

    // 256 WGs x 8 wave32 = 2048 waves = one 16-position M-tile per wave.
    dim3 grid(256), block(256);
    fired_wmma_kernel<<<grid, block, 0, stream>>>(x, rules, out);
}